// JointBlockDecoder_8907762172483
// MI455X (gfx1250) — compile-verified
//
#include <hip/hip_runtime.h>
#include <hip/hip_bf16.h>
#include <math.h>

typedef __attribute__((ext_vector_type(2))) float v2f;
typedef __attribute__((ext_vector_type(4))) float f4;
typedef __attribute__((ext_vector_type(8))) float v8f;

#define BDIM    8
#define NSPAN   65536
#define HDIM    256
#define WDIM    256
#define HWDIM   (HDIM * WDIM)
#define CDIM    64
#define EDIM    64
#define JITTER_SCALE 0.005859375f   /* 1.5/256 */

// ---------------------------------------------------------------------------
// Kernel 1: x[b,e,hw] = sum_c z[b,c,hw] * W_lat[e,c] + b_lat[e]
// stored as F[b][hw][e] (pixel-major, 64 contiguous floats per pixel) so the
// bilinear gather in kernel 2 reads contiguous 256B rows.
// One wave -> 64(e) x 16(pixel) strip via V_WMMA_F32_16X16X4_F32.
//   M = e (16 per tile, 4 tiles), N = pixel (16), K = c (4 per step, 16 steps)
// A (16x4 f32, 2 VGPR): lane l: m=l&15, k = 2*(l>>4) + {0,1}  -> b64 load
// B (4x16 f32, 2 VGPR): lane l: n=l&15, k = 2*(l>>4) + {0,1}  -> 2 b32 loads
// D (16x16 f32, 8 VGPR): lane l: n=l&15, m = v + 8*(l>>4)     -> contiguous
// ---------------------------------------------------------------------------
__global__ __launch_bounds__(256) void conv1x1_wmma_kernel(
    const float* __restrict__ z,      // [B][64][HW]
    const float* __restrict__ W_lat,  // [64][64]
    const float* __restrict__ b_lat,  // [64]
    float* __restrict__ F)            // [B][HW][64]
{
    const int wave = (int)((blockIdx.x * blockDim.x + threadIdx.x) >> 5);
    const int lane = (int)(threadIdx.x & 31);
    const int b       = wave >> 12;            // wave / (HW/16)
    const int hw_base = (wave & 4095) << 4;    // 16-pixel tile
    const int n    = lane & 15;
    const int half = lane >> 4;

    const float* zb = z + (size_t)b * CDIM * HWDIM + hw_base + n;

    v8f acc[4];
    #pragma unroll
    for (int t = 0; t < 4; ++t) {
        const float* bp = b_lat + t * 16 + 8 * half;
        #pragma unroll
        for (int v = 0; v < 8; ++v) acc[t][v] = bp[v];   // bias-init C
    }

    #pragma unroll
    for (int kk = 0; kk < 16; ++kk) {
        const int c0 = kk * 4 + 2 * half;
        v2f bm;
        bm.x = zb[(size_t)c0 * HWDIM];
        bm.y = zb[(size_t)(c0 + 1) * HWDIM];
        #pragma unroll
        for (int t = 0; t < 4; ++t) {
            const float* ap = W_lat + (size_t)(t * 16 + n) * CDIM + c0;
            v2f am;
            am.x = ap[0];
            am.y = ap[1];
            acc[t] = __builtin_amdgcn_wmma_f32_16x16x4_f32(
                false, am, false, bm, (short)0, acc[t], false, false);
        }
    }

    float* outp = F + ((size_t)b * HWDIM + hw_base + n) * EDIM;
    #pragma unroll
    for (int t = 0; t < 4; ++t) {
        float* o = outp + t * 16 + 8 * half;
        f4 lo = {acc[t][0], acc[t][1], acc[t][2], acc[t][3]};
        f4 hi = {acc[t][4], acc[t][5], acc[t][6], acc[t][7]};
        *(f4*)(o)     = lo;   // global_store_b128
        *(f4*)(o + 4) = hi;   // global_store_b128
    }
}

// ---------------------------------------------------------------------------
// Kernel 2: fused FiLM MLP + jitter + bilinear gather + 64->6 projection.
// One thread per (b, n) span. Weights staged in LDS (broadcast reads).
// ---------------------------------------------------------------------------
__global__ __launch_bounds__(256) void film_sample_kernel(
    const float* __restrict__ spans,       // [B][N][3]
    const float* __restrict__ emb,         // [23][8]
    const float* __restrict__ W1,          // [16][40]
    const float* __restrict__ b1,          // [16]
    const float* __restrict__ W2,          // [2][16]
    const float* __restrict__ b2,          // [2]
    const float* __restrict__ Wu,          // [6][64]
    const float* __restrict__ bu,          // [6]
    const float* __restrict__ freqs,       // [8]
    const float* __restrict__ base_coords, // [N][2]
    const float* __restrict__ F,           // [B][HW][64]
    float* __restrict__ out)               // [B][N][6]
{
    __shared__ float sW1[16 * 40];
    __shared__ float sWu[6 * 64];
    __shared__ float semb[23 * 8];
    __shared__ float sb1[16];
    __shared__ float sW2[2 * 16];
    __shared__ float sfr[8];
    __shared__ float sbu[6];
    __shared__ float sb2[2];

    for (int i = threadIdx.x; i < 16 * 40; i += 256) sW1[i]  = W1[i];
    for (int i = threadIdx.x; i < 6 * 64;  i += 256) sWu[i]  = Wu[i];
    for (int i = threadIdx.x; i < 23 * 8;  i += 256) semb[i] = emb[i];
    if (threadIdx.x < 32) sW2[threadIdx.x] = W2[threadIdx.x];
    if (threadIdx.x < 16) sb1[threadIdx.x] = b1[threadIdx.x];
    if (threadIdx.x < 8)  sfr[threadIdx.x] = freqs[threadIdx.x];
    if (threadIdx.x < 6)  sbu[threadIdx.x] = bu[threadIdx.x];
    if (threadIdx.x < 2)  sb2[threadIdx.x] = b2[threadIdx.x];
    __syncthreads();

    const size_t gid = (size_t)blockIdx.x * blockDim.x + threadIdx.x; // b*N + n
    const int b = (int)(gid >> 16);
    const int n = (int)(gid & 65535);

    const float* sp = spans + gid * 3;
    const int   chr   = (int)sp[0];            // matches astype(int32) for >=0
    const float start = sp[1];
    const float len   = sp[2];

    // Build FiLM input: [emb(8) | sin(2pi s f)(8) | cos(2pi s f)(8)
    //                    | sin(2pi l f)(8) | cos(2pi l f)(8)]
    float film[40];
    const float* e8 = semb + chr * 8;
    #pragma unroll
    for (int i = 0; i < 8; ++i) film[i] = e8[i];
    const float two_pi = 6.28318530717958647692f;
    #pragma unroll
    for (int f = 0; f < 8; ++f) {
        const float s = two_pi * start * sfr[f];
        const float l = two_pi * len   * sfr[f];
        film[8  + f] = __sinf(s);
        film[16 + f] = __cosf(s);
        film[24 + f] = __sinf(l);
        film[32 + f] = __cosf(l);
    }

    // h = leaky_relu(film @ W1^T + b1, 0.2)
    float h[16];
    #pragma unroll
    for (int i = 0; i < 16; ++i) {
        float a = sb1[i];
        const float* w = sW1 + i * 40;
        #pragma unroll
        for (int j = 0; j < 40; ++j) a = fmaf(w[j], film[j], a);
        h[i] = (a > 0.0f) ? a : 0.2f * a;
    }

    // delta = tanh(h @ W2^T + b2) * JITTER_SCALE
    float d[2];
    #pragma unroll
    for (int i = 0; i < 2; ++i) {
        float a = sb2[i];
        const float* w = sW2 + i * 16;
        #pragma unroll
        for (int j = 0; j < 16; ++j) a = fmaf(w[j], h[j], a);
        d[i] = tanhf(a) * JITTER_SCALE;
    }

    // uv = clip(base + delta, 0, 1); bilinear setup (identical corner math to ref)
    float u = base_coords[(size_t)n * 2 + 0] + d[0];
    float v = base_coords[(size_t)n * 2 + 1] + d[1];
    u = fminf(fmaxf(u, 0.0f), 1.0f);
    v = fminf(fmaxf(v, 0.0f), 1.0f);
    const float du = u * (float)(WDIM - 1);
    const float dv = v * (float)(HDIM - 1);
    const float j0f = floorf(du), i0f = floorf(dv);
    const int j0 = (int)j0f, i0 = (int)i0f;
    const int j1 = min(j0 + 1, WDIM - 1);
    const int i1 = min(i0 + 1, HDIM - 1);
    const float j1f = (float)j1, i1f = (float)i1;
    const float w00 = (i1f - dv) * (j1f - du);
    const float w10 = (i1f - dv) * (du - j0f);
    const float w01 = (dv - i0f) * (j1f - du);
    const float w11 = (dv - i0f) * (du - j0f);

    const float* Fb  = F + (size_t)b * HWDIM * EDIM;
    const float* p00 = Fb + (size_t)(i0 * WDIM + j0) * EDIM;
    const float* p10 = Fb + (size_t)(i0 * WDIM + j1) * EDIM;
    const float* p01 = Fb + (size_t)(i1 * WDIM + j0) * EDIM;
    const float* p11 = Fb + (size_t)(i1 * WDIM + j1) * EDIM;

    // g = bilinear(F); out = g @ Wu^T + bu, fused over 4-channel chunks
    float o[6];
    #pragma unroll
    for (int k = 0; k < 6; ++k) o[k] = sbu[k];

    #pragma unroll 4
    for (int e4 = 0; e4 < 16; ++e4) {
        const f4 a = *(const f4*)(p00 + e4 * 4);
        const f4 c = *(const f4*)(p10 + e4 * 4);
        const f4 e = *(const f4*)(p01 + e4 * 4);
        const f4 g = *(const f4*)(p11 + e4 * 4);
        f4 gv;
        gv.x = fmaf(w00, a.x, fmaf(w10, c.x, fmaf(w01, e.x, w11 * g.x)));
        gv.y = fmaf(w00, a.y, fmaf(w10, c.y, fmaf(w01, e.y, w11 * g.y)));
        gv.z = fmaf(w00, a.z, fmaf(w10, c.z, fmaf(w01, e.z, w11 * g.z)));
        gv.w = fmaf(w00, a.w, fmaf(w10, c.w, fmaf(w01, e.w, w11 * g.w)));
        #pragma unroll
        for (int k = 0; k < 6; ++k) {
            const float* wu = sWu + k * 64 + e4 * 4;
            o[k] = fmaf(gv.x, wu[0], o[k]);
            o[k] = fmaf(gv.y, wu[1], o[k]);
            o[k] = fmaf(gv.z, wu[2], o[k]);
            o[k] = fmaf(gv.w, wu[3], o[k]);
        }
    }

    float* op = out + gid * 6;
    #pragma unroll
    for (int k = 0; k < 6; ++k) op[k] = o[k];
}

// ---------------------------------------------------------------------------
extern "C" void kernel_launch(void* const* d_in, const int* in_sizes, int n_in,
                              void* d_out, int out_size, void* d_ws, size_t ws_size,
                              hipStream_t stream) {
    const float* z           = (const float*)d_in[0];
    const float* spans       = (const float*)d_in[1];
    const float* W_lat       = (const float*)d_in[2];
    const float* b_lat       = (const float*)d_in[3];
    const float* emb         = (const float*)d_in[4];
    const float* W1          = (const float*)d_in[5];
    const float* b1          = (const float*)d_in[6];
    const float* W2          = (const float*)d_in[7];
    const float* b2          = (const float*)d_in[8];
    const float* Wu          = (const float*)d_in[9];
    const float* bu          = (const float*)d_in[10];
    const float* freqs       = (const float*)d_in[11];
    const float* base_coords = (const float*)d_in[12];
    float* out = (float*)d_out;
    float* F   = (float*)d_ws;   // needs B*HW*E*4 = 128 MiB scratch

    // Kernel 1: 8 batches * (65536/16) pixel tiles / 8 waves-per-block
    conv1x1_wmma_kernel<<<dim3(4096), dim3(256), 0, stream>>>(z, W_lat, b_lat, F);

    // Kernel 2: one thread per span: 8*65536/256 blocks
    film_sample_kernel<<<dim3(2048), dim3(256), 0, stream>>>(
        spans, emb, W1, b1, W2, b2, Wu, bu, freqs, base_coords, F, out);
}